// Projector_32762010534303
// MI455X (gfx1250) — compile-verified
//
#include <hip/hip_runtime.h>
#include <hip/hip_bf16.h>

typedef __attribute__((ext_vector_type(2))) float v2f;
typedef __attribute__((ext_vector_type(8))) float v8f;

#define N_CIN   64
#define N_CHID  16
#define N_COUT  64
#define BDIM    4
#define HDIM    352
#define WDIM    1216
#define HW_     (HDIM * WDIM)
#define NPIX    (BDIM * HW_)
#define EPS_    1e-5f
#define STATS_BLOCKS 256
#define STATS_WAVES  (STATS_BLOCKS * 8)

// ---------------------------------------------------------------------------
// Kernel 1: MLP 64 -> relu(16) -> 64 via V_WMMA_F32_16X16X4_F32.
// One wave32 handles one 16-point tile. Layer-1 hidden tile staged through
// LDS (stride 17 to dodge bank conflicts) to convert C-layout -> A-layout.
// ---------------------------------------------------------------------------
__global__ __launch_bounds__(256) void mlp_wmma_kernel(
    const float* __restrict__ X,   // [N, 64]
    const float* __restrict__ W1,  // [64, 16]
    const float* __restrict__ b1,  // [16]
    const float* __restrict__ W2,  // [16, 64]
    const float* __restrict__ b2,  // [64]
    float* __restrict__ feats,     // [N, 64]
    int N, int ntiles)
{
    __shared__ float hbuf[8 * 16 * 17];
    const int lane = threadIdx.x & 31;
    const int wave = threadIdx.x >> 5;
    const int col  = lane & 15;   // N-index for B/C; M-index (row) for A
    const int half = lane >> 4;   // selects K sub-pair for A/B operands
    float* hl = &hbuf[wave * 16 * 17];

    int gw     = blockIdx.x * 8 + wave;
    int stride = gridDim.x * 8;

    for (int t = gw; t < ntiles; t += stride) {
        const int row0 = t * 16;
        int m = row0 + col;              // point row this lane feeds into A
        if (m >= N) m = N - 1;           // clamp tail (N is a multiple of 16 here)
        const float* xr = X + (size_t)m * N_CIN;

        // ----- layer 1: C[16x16] = X_tile[16x64] @ W1[64x16], K in steps of 4
        v8f acc = {0, 0, 0, 0, 0, 0, 0, 0};
        #pragma unroll
        for (int k0 = 0; k0 < N_CIN; k0 += 4) {
            const int k = k0 + 2 * half;
            v2f a; a.x = xr[k];                    a.y = xr[k + 1];
            v2f b; b.x = W1[k * N_CHID + col];     b.y = W1[(k + 1) * N_CHID + col];
            acc = __builtin_amdgcn_wmma_f32_16x16x4_f32(
                false, a, false, b, (short)0, acc, false, false);
        }
        const float bias1 = b1[col];
        #pragma unroll
        for (int r = 0; r < 8; ++r) {
            float h = acc[r] + bias1;
            h = h > 0.0f ? h : 0.0f;               // relu
            hl[(r + 8 * half) * 17 + col] = h;     // C-layout -> LDS (row-major 16x16)
        }

        // ----- layer 2: out[16x64] = H[16x16] @ W2[16x64], four 16-col tiles
        #pragma unroll
        for (int j = 0; j < 4; ++j) {
            v8f acc2 = {0, 0, 0, 0, 0, 0, 0, 0};
            #pragma unroll
            for (int k0 = 0; k0 < N_CHID; k0 += 4) {
                const int k = k0 + 2 * half;
                v2f a; a.x = hl[col * 17 + k];     a.y = hl[col * 17 + k + 1];
                v2f b; b.x = W2[k * N_COUT + j * 16 + col];
                       b.y = W2[(k + 1) * N_COUT + j * 16 + col];
                acc2 = __builtin_amdgcn_wmma_f32_16x16x4_f32(
                    false, a, false, b, (short)0, acc2, false, false);
            }
            const float bias2 = b2[j * 16 + col];
            #pragma unroll
            for (int r = 0; r < 8; ++r) {
                const int rowg = row0 + r + 8 * half;
                if (rowg < N)
                    feats[(size_t)rowg * N_COUT + j * 16 + col] = acc2[r] + bias2;
            }
        }
    }
}

// ---------------------------------------------------------------------------
// Kernel 2: winner grid init to -1
// ---------------------------------------------------------------------------
__global__ __launch_bounds__(256) void init_winner_kernel(int* __restrict__ winner)
{
    int pix = blockIdx.x * 256 + threadIdx.x;
    if (pix < NPIX) winner[pix] = -1;
}

// ---------------------------------------------------------------------------
// Kernel 3: pinhole projection + deterministic scatter (highest index wins)
// ---------------------------------------------------------------------------
__global__ __launch_bounds__(256) void project_kernel(
    const float* __restrict__ pos,   // [N, 3]
    const int*   __restrict__ batch, // [N]
    const float* __restrict__ cam,   // [B, 3, 3]
    int* __restrict__ winner,        // [NPIX]
    int* __restrict__ flatbuf,       // [N]
    int N)
{
    int i = blockIdx.x * 256 + threadIdx.x;
    if (i >= N) return;
    const int b = batch[i];
    const float* K = cam + b * 9;
    const float x = pos[i * 3 + 0], y = pos[i * 3 + 1], z = pos[i * 3 + 2];
    const float px = K[0] * x + K[1] * y + K[2] * z;
    const float py = K[3] * x + K[4] * y + K[5] * z;
    const float pz = K[6] * x + K[7] * y + K[8] * z;
    const float u = px / pz;
    const float v = py / pz;
    const int ui = (int)fminf(fmaxf(floorf(u), 0.0f), (float)(WDIM - 1));
    const int vi = (int)fminf(fmaxf(floorf(v), 0.0f), (float)(HDIM - 1));
    const int flat = b * HW_ + vi * WDIM + ui;
    flatbuf[i] = flat;
    atomicMax(&winner[flat], i);
}

// ---------------------------------------------------------------------------
// Kernel 4: per-wave partial channel stats over winning points.
// Lane L accumulates channels L and L+32 -> deterministic tree reduction.
// Partial row layout: [0:64) sum_c, [64:128) sumsq_c, [128] count.
// ---------------------------------------------------------------------------
__global__ __launch_bounds__(256) void stats_kernel(
    const float* __restrict__ feats,
    const int*   __restrict__ flatbuf,
    const int*   __restrict__ winner,
    float* __restrict__ partials,    // [STATS_WAVES, 132]
    int N)
{
    const int lane = threadIdx.x & 31;
    const int gw   = (blockIdx.x * 256 + threadIdx.x) >> 5;
    const int nw   = gridDim.x * 8;
    float s0 = 0.f, s1 = 0.f, q0 = 0.f, q1 = 0.f, cnt = 0.f;
    for (int i = gw; i < N; i += nw) {
        const int flat = flatbuf[i];          // uniform per wave -> broadcast load
        if (winner[flat] == i) {
            const float* f = feats + (size_t)i * 64;
            const float a = f[lane];
            const float b = f[lane + 32];
            s0 += a; q0 += a * a;
            s1 += b; q1 += b * b;
            cnt += 1.0f;
        }
    }
    float* row = partials + (size_t)gw * 132;
    row[lane]      = s0;
    row[32 + lane] = s1;
    row[64 + lane] = q0;
    row[96 + lane] = q1;
    if (lane == 0) row[128] = cnt;
}

// ---------------------------------------------------------------------------
// Kernel 5: fold partials, add analytic zero_encoding contribution, emit
// per-channel affine: ab[c] = gamma*rsqrt(var+eps), ab[64+c] = beta - mean*ab[c]
// ---------------------------------------------------------------------------
__global__ __launch_bounds__(256) void reduce_kernel(
    const float* __restrict__ partials,
    const float* __restrict__ zero_enc,
    const float* __restrict__ gamma,
    const float* __restrict__ beta,
    float* __restrict__ ab,          // [128]
    int nrows)
{
    __shared__ float cols[129];
    const int tid = threadIdx.x;
    if (tid < 129) {
        float s = 0.0f;
        for (int r = 0; r < nrows; ++r)
            s += partials[(size_t)r * 132 + tid];
        cols[tid] = s;
    }
    __syncthreads();
    if (tid < 64) {
        const float P    = (float)NPIX;
        const float nset = cols[128];
        const float z    = zero_enc[tid];
        const float S    = cols[tid]      + z * (P - nset);
        const float Q    = cols[64 + tid] + z * z * (P - nset);
        const float mean = S / P;
        const float var  = Q / P - mean * mean;
        const float a    = gamma[tid] * rsqrtf(var + EPS_);
        ab[tid]      = a;
        ab[64 + tid] = beta[tid] - mean * a;
    }
}

// ---------------------------------------------------------------------------
// Kernel 6: fused scatter-gather + batchnorm. One thread per pixel; per
// channel the wave stores 32 consecutive floats (coalesced plane writes).
// ---------------------------------------------------------------------------
__global__ __launch_bounds__(256) void scatter_norm_kernel(
    const float* __restrict__ feats,
    const int*   __restrict__ winner,
    const float* __restrict__ zero_enc,
    const float* __restrict__ ab,
    float* __restrict__ out)         // [B, 64, H, W]
{
    const int pix = blockIdx.x * 256 + threadIdx.x;
    if (pix >= NPIX) return;
    const int b   = pix / HW_;
    const int rem = pix - b * HW_;
    const int w   = winner[pix];
    const float* src = (w >= 0) ? (feats + (size_t)w * 64) : zero_enc;
    float* o = out + (size_t)b * 64 * HW_ + rem;
    #pragma unroll 4
    for (int c = 0; c < 64; ++c) {
        o[(size_t)c * HW_] = src[c] * ab[c] + ab[64 + c];
    }
}

// ---------------------------------------------------------------------------
extern "C" void kernel_launch(void* const* d_in, const int* in_sizes, int n_in,
                              void* d_out, int out_size, void* d_ws, size_t ws_size,
                              hipStream_t stream)
{
    (void)n_in; (void)out_size; (void)ws_size;
    const float* pc_features = (const float*)d_in[0];
    const float* pc_pos      = (const float*)d_in[1];
    const int*   pc_batch    = (const int*)d_in[2];
    const float* cam         = (const float*)d_in[3];
    const float* W1          = (const float*)d_in[4];
    const float* b1          = (const float*)d_in[5];
    const float* W2          = (const float*)d_in[6];
    const float* b2          = (const float*)d_in[7];
    const float* zero_enc    = (const float*)d_in[8];
    const float* gamma       = (const float*)d_in[9];
    const float* beta        = (const float*)d_in[10];
    const int N = in_sizes[2];           // pc_batch element count = point count
    float* out = (float*)d_out;

    // workspace carve-up (all chunks 16B-aligned)
    char* ws = (char*)d_ws;
    float* feats    = (float*)ws;  ws += (size_t)N * 64 * sizeof(float);
    int*   winner   = (int*)ws;    ws += (size_t)NPIX * sizeof(int);
    int*   flatbuf  = (int*)ws;    ws += (size_t)N * sizeof(int);
    float* partials = (float*)ws;  ws += (size_t)STATS_WAVES * 132 * sizeof(float);
    float* ab       = (float*)ws;

    const int ntiles = (N + 15) / 16;
    mlp_wmma_kernel<<<(ntiles + 7) / 8, 256, 0, stream>>>(
        pc_features, W1, b1, W2, b2, feats, N, ntiles);
    init_winner_kernel<<<(NPIX + 255) / 256, 256, 0, stream>>>(winner);
    project_kernel<<<(N + 255) / 256, 256, 0, stream>>>(
        pc_pos, pc_batch, cam, winner, flatbuf, N);
    stats_kernel<<<STATS_BLOCKS, 256, 0, stream>>>(
        feats, flatbuf, winner, partials, N);
    reduce_kernel<<<1, 256, 0, stream>>>(
        partials, zero_enc, gamma, beta, ab, STATS_WAVES);
    scatter_norm_kernel<<<(NPIX + 255) / 256, 256, 0, stream>>>(
        feats, winner, zero_enc, ab, out);
}